// WorldToLocalEncoder_28355374088625
// MI455X (gfx1250) — compile-verified
//
#include <hip/hip_runtime.h>

typedef float v2f __attribute__((ext_vector_type(2)));
typedef float v8f __attribute__((ext_vector_type(8)));

// Problem geometry from the reference
#define SAMPLES   96
#define NJOINTS   24
#define MROWS     (NJOINTS * 4)      // 96 rows, m = j*4 + a (a==3 discarded)
#define S_TILES   (SAMPLES / 16)     // 6
#define M_TILES   (MROWS / 16)       // 6
#define WAVES_PER_BLOCK 8

__global__ __launch_bounds__(256)
void w2l_wmma_f32_16x16x4(const float* __restrict__ pts,   // [R,96,3]
                          const float* __restrict__ skts,  // [R,24,4,4]
                          float* __restrict__ out)         // [R,96,24,3]
{
    const int lane = threadIdx.x & 31;          // wave32
    const int wid  = blockIdx.x * WAVES_PER_BLOCK + (threadIdx.x >> 5);
    const int r    = wid / S_TILES;             // ray
    const int st   = wid - r * S_TILES;         // sample tile
    const int kh   = lane >> 4;                 // 0: K={0,1}, 1: K={2,3}
    const int l    = lane & 15;
    const int s    = st * 16 + l;               // this lane's sample (B column)

    // ---- B matrix 4x16 (KxN), 2 VGPRs ----
    // lower lanes: (x, y); upper lanes: (z, 1.0)  -- branchless, EXEC uniform
    const float* prow = pts + (size_t)r * (SAMPLES * 3) + (size_t)s * 3;
    v2f bm;
    bm.x = prow[kh * 2];                        // x or z
    bm.y = kh ? 1.0f : prow[1];                 // y or homogeneous 1.0

    const float* sk   = skts + (size_t)r * (MROWS * 4);
    float*       orow = out  + (size_t)r * (SAMPLES * NJOINTS * 3)
                             + (size_t)s * (NJOINTS * 3);

#pragma unroll
    for (int mt = 0; mt < M_TILES; ++mt) {
        // ---- A matrix 16x4 (MxK), 2 VGPRs: one 8B load per lane ----
        const int m = mt * 16 + l;              // A row for this lane
        v2f am = *(const v2f*)(sk + m * 4 + kh * 2);

        v8f c = {};
        // D = A x B + 0   (single WMMA covers the whole K=4 reduction)
        v8f d = __builtin_amdgcn_wmma_f32_16x16x4_f32(
            /*neg_a=*/false, am, /*neg_b=*/false, bm,
            /*c_mod=*/(short)0, c, /*reuse_a=*/false, /*reuse_b=*/false);

        // D layout: VGPR v, this lane holds row m_row = mt*16 + kh*8 + v.
        // a = m_row & 3 = v & 3 (compile-time): skip v==3 and v==7 (a==3).
        const int jbase = (mt * 16 + kh * 8) >> 2;   // joint of v=0
        float* o0 = orow + jbase * 3;
        o0[0] = d[0]; o0[1] = d[1]; o0[2] = d[2];    // j=jbase,   a=0..2
        o0[3] = d[4]; o0[4] = d[5]; o0[5] = d[6];    // j=jbase+1, a=0..2
    }
}

extern "C" void kernel_launch(void* const* d_in, const int* in_sizes, int n_in,
                              void* d_out, int out_size, void* d_ws, size_t ws_size,
                              hipStream_t stream) {
    const float* pts  = (const float*)d_in[0];   // [R,96,3] fp32
    const float* skts = (const float*)d_in[1];   // [R,24,4,4] fp32
    float* out = (float*)d_out;                  // [R,96,24,3] fp32

    const int R = in_sizes[0] / (SAMPLES * 3);   // 4096
    const int total_waves = R * S_TILES;         // one wave per (ray, s-tile)
    const int blocks = (total_waves + WAVES_PER_BLOCK - 1) / WAVES_PER_BLOCK;

    w2l_wmma_f32_16x16x4<<<blocks, WAVES_PER_BLOCK * 32, 0, stream>>>(pts, skts, out);
}